// SnakeFastLoss_37331855737016
// MI455X (gfx1250) — compile-verified
//
#include <hip/hip_runtime.h>
#include <math.h>

// Problem constants (match reference)
#define B_      4
#define H_      256
#define W_      256
#define K_      128
#define STEPSZ_ 0.1f
#define ALPHA_  0.01f
#define BETA_   0.01f
#define EXTG_   10.0f
#define NSTEPS_ 20
#define NSTEPSW_ 10
#define DMAX_   20.0f
#define NEG_W_  1.5f
#define ENH_    10.0f
#define FW_     13   // filter width (stdev=2 -> r=6 -> 13)
#define FR_     6

typedef __attribute__((ext_vector_type(2))) float v2f;
typedef __attribute__((ext_vector_type(8))) float v8f;

// fast hardware sqrt: lowers to a single v_sqrt_f32 (~1 ulp), avoiding the
// correctly-rounded ~20-instruction expansion that dominated the VALU count.
__device__ __forceinline__ float fsqrt_hw(float x) {
    return __builtin_amdgcn_sqrtf(x);
}

// ---------- ordered-uint keys for float atomic min/max ----------
__device__ __forceinline__ unsigned fkey(float f) {
    unsigned u = __float_as_uint(f);
    return (u & 0x80000000u) ? ~u : (u ^ 0x80000000u);
}
__device__ __forceinline__ float fromkey(unsigned k) {
    unsigned u = (k & 0x80000000u) ? (k ^ 0x80000000u) : ~k;
    return __uint_as_float(u);
}

// ---------- on-the-fly 13x13 cross-correlation at one pixel ----------
// fl points to LDS copy of filter, layout [2][13][13] (channel 0 = d/dy, 1 = d/dx)
__device__ __forceinline__ void conv2_at(const float* __restrict__ img,
                                         const float* __restrict__ fl,
                                         int yy, int xx, bool useAbs,
                                         float& oy, float& ox) {
    float ay = 0.f, ax = 0.f;
    for (int dy = 0; dy < FW_; ++dy) {
        int sy = yy + dy - FR_;
        if ((unsigned)sy >= (unsigned)H_) continue;
        const float* row = img + sy * W_;
        const float* fy = fl + dy * FW_;
        const float* fx = fl + FW_ * FW_ + dy * FW_;
        for (int dx = 0; dx < FW_; ++dx) {
            int sx = xx + dx - FR_;
            if ((unsigned)sx >= (unsigned)W_) continue;
            float v = row[sx];
            if (useAbs) v = fabsf(v);
            ay = fmaf(fy[dx], v, ay);
            ax = fmaf(fx[dx], v, ax);
        }
    }
    oy = ay * EXTG_;
    ox = ax * EXTG_;
}

// bilinear sample of the (on-the-fly) gradient image at (py,px)
__device__ __forceinline__ void sample_grad(const float* __restrict__ img,
                                            const float* __restrict__ fl,
                                            float py, float px, bool useAbs,
                                            float& g0, float& g1) {
    float y = fminf(fmaxf(py, 0.f), (float)H_ - 1.001f);
    float x = fminf(fmaxf(px, 0.f), (float)W_ - 1.001f);
    int y0 = (int)y, x0 = (int)x;       // floor (values are non-negative)
    float wy = y - (float)y0, wx = x - (float)x0;
    float c00y, c00x, c01y, c01x, c10y, c10x, c11y, c11x;
    conv2_at(img, fl, y0,     x0,     useAbs, c00y, c00x);
    conv2_at(img, fl, y0,     x0 + 1, useAbs, c01y, c01x);
    conv2_at(img, fl, y0 + 1, x0,     useAbs, c10y, c10x);
    conv2_at(img, fl, y0 + 1, x0 + 1, useAbs, c11y, c11x);
    float w00 = (1.f - wy) * (1.f - wx), w01 = (1.f - wy) * wx;
    float w10 = wy * (1.f - wx),         w11 = wy * wx;
    g0 = c00y * w00 + c01y * w01 + c10y * w10 + c11y * w11;
    g1 = c00x * w00 + c01x * w01 + c10x * w10 + c11x * w11;
}

// ---------- kernel 0: init accumulators ----------
__global__ void init_kernel(unsigned* minkey, unsigned* maxkey, float* acc) {
    *minkey = 0xFFFFFFFFu;  // key of +inf
    *maxkey = 0u;           // key of -inf
    *acc = 0.f;
}

// ---------- kernel 1: snake + width evolution (1 block/batch, 1 thread/node) ----------
__global__ __launch_bounds__(K_) void snake_kernel(
        const float* __restrict__ pred, const float* __restrict__ node_pos,
        const float* __restrict__ widths, const float* __restrict__ fltr,
        float* __restrict__ posy, float* __restrict__ posx, float* __restrict__ widout) {
    __shared__ float s_fl[2 * FW_ * FW_];
    __shared__ float s_py[K_], s_px[K_];
    const int k = threadIdx.x;
    const int b = blockIdx.x;
    const float* img = pred + b * H_ * W_;

    for (int i = k; i < 2 * FW_ * FW_; i += K_) s_fl[i] = fltr[i];
    float py = node_pos[(b * K_ + k) * 2 + 0];
    float px = node_pos[(b * K_ + k) * 2 + 1];
    s_py[k] = py; s_px[k] = px;
    __syncthreads();

    const int km1 = (k + K_ - 1) & (K_ - 1), kp1 = (k + 1) & (K_ - 1);
    const int km2 = (k + K_ - 2) & (K_ - 1), kp2 = (k + 2) & (K_ - 1);

    for (int step = 0; step < NSTEPS_; ++step) {
        float pym1 = s_py[km1], pyp1 = s_py[kp1], pym2 = s_py[km2], pyp2 = s_py[kp2];
        float pxm1 = s_px[km1], pxp1 = s_px[kp1], pxm2 = s_px[km2], pxp2 = s_px[kp2];
        float fy, fx;
        sample_grad(img, s_fl, py, px, false, fy, fx);
        float d2y = pyp1 - 2.f * py + pym1;
        float d2x = pxp1 - 2.f * px + pxm1;
        float d4y = pyp2 - 4.f * pyp1 + 6.f * py - 4.f * pym1 + pym2;
        float d4x = pxp2 - 4.f * pxp1 + 6.f * px - 4.f * pxm1 + pxm2;
        py += STEPSZ_ * (ALPHA_ * d2y - BETA_ * d4y + fy);
        px += STEPSZ_ * (ALPHA_ * d2x - BETA_ * d4x + fx);
        __syncthreads();           // all reads done
        s_py[k] = py; s_px[k] = px;
        __syncthreads();           // all writes visible
    }

    // tangent / normal
    float ty = s_py[kp1] - s_py[km1];
    float tx = s_px[kp1] - s_px[km1];
    float nl = fsqrt_hw(ty * ty + tx * tx) + 1e-6f;
    ty /= nl; tx /= nl;
    float n0 = -tx, n1 = ty;

    // width evolution: gradient of |img| sampled at final (fixed) pos -> dw constant
    float g0, g1;
    sample_grad(img, s_fl, py, px, true, g0, g1);
    float dw = g0 * n0 + g1 * n1;
    float wv = widths[b * K_ + k];
    for (int s = 0; s < NSTEPSW_; ++s) wv = fmaxf(wv + STEPSZ_ * dw, 0.f);

    posy[b * K_ + k] = py;
    posx[b * K_ + k] = px;
    widout[b * K_ + k] = wv;
}

// ---------- kernel 2: distance map via f32 WMMA (rank-4 squared-distance trick) ----------
// A (16 nodes x 4) row m = [-2*py, -2*px, 1, py^2+px^2]
// B (4 x 16 pixels) col n = [y, x, y^2+x^2, 1]   =>  D[m][n] = (y-py)^2 + (x-px)^2
// f32 16x16x4 layouts (ISA 7.12.2):
//   A: lanes 0-15 hold K=0,1 in v0,v1 for row M=lane; lanes 16-31 hold K=2,3.
//   B: lanes 0-15 hold col N=lane for K=0,1; lanes 16-31 for K=2,3 (mirror of A).
//   D: vgpr j = row (M%8), lanes 0-15 -> M=j, lanes 16-31 -> M=j+8; N = lane%16.
__global__ __launch_bounds__(256) void dmap_kernel(
        const float* __restrict__ posy, const float* __restrict__ posx,
        const float* __restrict__ widv, float* __restrict__ snake_dm,
        unsigned* __restrict__ minkey, unsigned* __restrict__ maxkey) {
    __shared__ float s_py[K_], s_px[K_], s_w[K_];
    __shared__ float s_red[16];
    const int b = blockIdx.y;
    const int t = threadIdx.x;
    if (t < K_) {
        s_py[t] = posy[b * K_ + t];
        s_px[t] = posx[b * K_ + t];
        s_w[t]  = widv[b * K_ + t];
    }
    __syncthreads();

    const int wave = t >> 5, lane = t & 31;
    const int lh = lane & 15;
    const bool hi = lane >= 16;
    const int y = blockIdx.x * 8 + wave;   // one image row per wave

    // A operands for the 8 node groups (16 nodes each), kept in registers
    float a0[8], a1[8];
#pragma unroll
    for (int g = 0; g < 8; ++g) {
        float py = s_py[g * 16 + lh], px = s_px[g * 16 + lh];
        a0[g] = hi ? 1.0f : (-2.0f * py);
        a1[g] = hi ? fmaf(py, py, px * px) : (-2.0f * px);
    }

    const float fy = (float)y;
    float rowmin = 1e30f, rowmax = -1e30f;

    for (int tile = 0; tile < 16; ++tile) {
        float fx = (float)(tile * 16 + lh);
        v2f bvec;
        bvec.x = hi ? fmaf(fy, fy, fx * fx) : fy;
        bvec.y = hi ? 1.0f : fx;

        float m = 1e30f;
#pragma unroll
        for (int g = 0; g < 8; ++g) {
            v2f avec; avec.x = a0[g]; avec.y = a1[g];
            v8f c = {};
            c = __builtin_amdgcn_wmma_f32_16x16x4_f32(
                    false, avec, false, bvec, (short)0, c, false, false);
            const int base = g * 16 + (hi ? 8 : 0);
#pragma unroll
            for (int j = 0; j < 8; ++j) {
                float d = fsqrt_hw(c[j] + 1e-8f) - s_w[base + j];
                m = fminf(m, d);
            }
        }
        // combine node rows 0-7 (lanes<16) with rows 8-15 (lanes>=16)
        float other = __shfl_xor(m, 16, 32);
        float dmv = fminf(fminf(m, other), DMAX_);
        if (!hi) snake_dm[(b * H_ + y) * W_ + tile * 16 + lh] = dmv;
        rowmin = fminf(rowmin, dmv);
        rowmax = fmaxf(rowmax, dmv);
    }

    // wave-level min/max (duplicates in upper half are harmless)
#pragma unroll
    for (int off = 8; off >= 1; off >>= 1) {
        rowmin = fminf(rowmin, __shfl_xor(rowmin, off, 32));
        rowmax = fmaxf(rowmax, __shfl_xor(rowmax, off, 32));
    }
    if (lane == 0) { s_red[wave] = rowmin; s_red[8 + wave] = rowmax; }
    __syncthreads();
    if (t == 0) {
        float mn = s_red[0], mx = s_red[8];
        for (int w2 = 1; w2 < 8; ++w2) { mn = fminf(mn, s_red[w2]); mx = fmaxf(mx, s_red[8 + w2]); }
        atomicMin(minkey, fkey(mn));
        atomicMax(maxkey, fkey(mx));
    }
}

// ---------- kernel 3: weighted MSE partial sums ----------
__global__ __launch_bounds__(256) void loss_kernel(
        const float* __restrict__ pred, const float* __restrict__ snake_dm,
        const unsigned* __restrict__ minkey, const unsigned* __restrict__ maxkey,
        float* __restrict__ acc) {
    __shared__ float s_part[8];
    const int idx = blockIdx.x * 256 + threadIdx.x;
    const float mn = fromkey(*minkey), mx = fromkey(*maxkey);
    const bool enhance = ((mx - mn) < 30.0f) && (mn > -10.0f);

    float sdm = snake_dm[idx];
    if (enhance && sdm < 0.f) sdm *= ENH_;
    float diff = pred[idx] - sdm;
    float w = (sdm < 0.f) ? NEG_W_ : 1.0f;
    float v = diff * diff * w;

#pragma unroll
    for (int off = 16; off >= 1; off >>= 1) v += __shfl_xor(v, off, 32);
    const int lane = threadIdx.x & 31, wave = threadIdx.x >> 5;
    if (lane == 0) s_part[wave] = v;
    __syncthreads();
    if (threadIdx.x == 0) {
        float s = 0.f;
        for (int w2 = 0; w2 < 8; ++w2) s += s_part[w2];
        atomicAdd(acc, s);
    }
}

// ---------- kernel 4: finalize ----------
__global__ void finalize_kernel(const float* __restrict__ acc, float* __restrict__ out) {
    out[0] = acc[0] * (1.0f / (float)(B_ * H_ * W_));
}

extern "C" void kernel_launch(void* const* d_in, const int* in_sizes, int n_in,
                              void* d_out, int out_size, void* d_ws, size_t ws_size,
                              hipStream_t stream) {
    const float* pred     = (const float*)d_in[0];   // (B,1,H,W)
    const float* node_pos = (const float*)d_in[1];   // (B,K,2)
    const float* widths   = (const float*)d_in[2];   // (B,K)
    const float* fltr     = (const float*)d_in[3];   // (2,1,13,13)
    float* out = (float*)d_out;

    // workspace layout (needs ~1.01 MB)
    float*    posy   = (float*)d_ws;                       // B*K
    float*    posx   = posy + B_ * K_;                     // B*K
    float*    widw   = posx + B_ * K_;                     // B*K
    unsigned* minkey = (unsigned*)(widw + B_ * K_);
    unsigned* maxkey = minkey + 1;
    float*    acc    = (float*)(maxkey + 1);
    float*    sdm    = (float*)((char*)d_ws + 8192);       // B*H*W floats

    init_kernel<<<1, 1, 0, stream>>>(minkey, maxkey, acc);
    snake_kernel<<<dim3(B_), K_, 0, stream>>>(pred, node_pos, widths, fltr,
                                              posy, posx, widw);
    dmap_kernel<<<dim3(H_ / 8, B_), 256, 0, stream>>>(posy, posx, widw,
                                                      sdm, minkey, maxkey);
    loss_kernel<<<dim3((B_ * H_ * W_) / 256), 256, 0, stream>>>(pred, sdm,
                                                                minkey, maxkey, acc);
    finalize_kernel<<<1, 1, 0, stream>>>(acc, out);
}